// StaticCacheAttention_39779987096356
// MI455X (gfx1250) — compile-verified
//
#include <hip/hip_runtime.h>
#include <hip/hip_bf16.h>

// ---------------- problem constants ----------------
#define BB   4
#define TT   16
#define HID  2048
#define HH   16
#define DD   128
#define SMAX 4096
#define PAST 2048
#define SS   (PAST + TT)          // 2064
#define MM   (BB * TT)            // 64 rows in all dense GEMMs

typedef __attribute__((ext_vector_type(2))) float v2f;
typedef __attribute__((ext_vector_type(8))) float v8f;

// strides inside the cache tensors (B,H,SMAX,D)
#define C_BSTR ((size_t)HH * SMAX * DD)   // 8388608
#define C_HSTR ((size_t)SMAX * DD)        // 524288

// ---------------------------------------------------------------------------
// Generic Y(M=64,N=2048) = X(64,K) @ W^T, W is (N,K) row-major (torch Linear).
// One wave per 16x16 output tile, fp32 WMMA, K-loop of 16x16x4 steps.
// A frag: lane holds X[mt*16 + lane%16][k0 + 2*(lane/16) + i]   (contig float2)
// B frag: lane holds W[nt*16 + lane%16][k0 + 2*(lane/16) + i]   (contig float2)
// ---------------------------------------------------------------------------
__global__ void gemm_xwt_wmma(const float* __restrict__ X,
                              const float* __restrict__ W,
                              float* __restrict__ Y,
                              int K, int N) {
  const int gtid = blockIdx.x * blockDim.x + threadIdx.x;
  const int wave = gtid >> 5;
  const int lane = gtid & 31;
  const int ntiles = N >> 4;
  const int mt = wave / ntiles;
  const int nt = wave % ntiles;
  if (mt >= (MM >> 4)) return;

  const int half = lane >> 4;   // which K pair this half-wave owns
  const int lm   = lane & 15;

  const float* __restrict__ xrow = X + (size_t)(mt * 16 + lm) * K + half * 2;
  const float* __restrict__ wrow = W + (size_t)(nt * 16 + lm) * K + half * 2;

  v8f acc = {};
  for (int k0 = 0; k0 < K; k0 += 4) {
    v2f a = *(const v2f*)(xrow + k0);
    v2f b = *(const v2f*)(wrow + k0);
    acc = __builtin_amdgcn_wmma_f32_16x16x4_f32(false, a, false, b,
                                                (short)0, acc, false, false);
  }
#pragma unroll
  for (int r = 0; r < 8; ++r) {
    Y[(size_t)(mt * 16 + r + 8 * half) * N + nt * 16 + lm] = acc[r];
  }
}

// ---------------------------------------------------------------------------
// Scatter the 16 freshly projected K/V rows into cache rows [PAST, PAST+T).
// k2d/v2d layout: [(b*T+t)][h*128+d]
// ---------------------------------------------------------------------------
__global__ void scatter_kv(const float* __restrict__ kp,
                           const float* __restrict__ vp,
                           float* __restrict__ newK,
                           float* __restrict__ newV) {
  const int n = blockIdx.x * blockDim.x + threadIdx.x;
  if (n >= MM * HID) return;
  const int d = n & (DD - 1);
  const int h = (n >> 7) & (HH - 1);
  const int t = (n >> 11) & (TT - 1);
  const int b = n >> 15;
  const size_t dst = (size_t)b * C_BSTR + (size_t)h * C_HSTR +
                     (size_t)(PAST + t) * DD + d;
  newK[dst] = kp[n];
  newV[dst] = vp[n];
}

// ---------------------------------------------------------------------------
// scores[bh][t][s] = scale * sum_d Q[b,h,t,d] * K'[b,h,s,d]
// One wave per (b,h, s-tile); 16x16 tile, K-loop over D=128.
// ---------------------------------------------------------------------------
__global__ void qk_scores_wmma(const float* __restrict__ Q2d,
                               const float* __restrict__ newK,
                               float* __restrict__ scores) {
  const int gtid = blockIdx.x * blockDim.x + threadIdx.x;
  const int wave = gtid >> 5;
  const int lane = gtid & 31;
  const int ST = SS >> 4;                 // 129 s-tiles
  const int st = wave % ST;
  const int bh = wave / ST;
  if (bh >= BB * HH) return;
  const int b = bh >> 4, h = bh & 15;
  const int half = lane >> 4, lm = lane & 15;

  const float* __restrict__ qrow =
      Q2d + (size_t)(b * TT + lm) * HID + h * DD + half * 2;
  const float* __restrict__ krow =
      newK + (size_t)b * C_BSTR + (size_t)h * C_HSTR +
      (size_t)(st * 16 + lm) * DD + half * 2;

  v8f acc = {};
  for (int k0 = 0; k0 < DD; k0 += 4) {
    v2f a  = *(const v2f*)(qrow + k0);
    v2f bb = *(const v2f*)(krow + k0);
    acc = __builtin_amdgcn_wmma_f32_16x16x4_f32(false, a, false, bb,
                                                (short)0, acc, false, false);
  }
  const float sc = 0.08838834764831845f;  // 1/sqrt(128)
  float* __restrict__ srow = scores + (size_t)bh * TT * SS;
#pragma unroll
  for (int r = 0; r < 8; ++r) {
    srow[(size_t)(r + 8 * half) * SS + st * 16 + lm] = acc[r] * sc;
  }
}

// ---------------------------------------------------------------------------
// In-place row softmax over 1024 rows of length 2064; one wave32 per row.
// ---------------------------------------------------------------------------
__global__ void softmax_rows(float* __restrict__ scores) {
  const int gtid = blockIdx.x * blockDim.x + threadIdx.x;
  const int wave = gtid >> 5;
  const int lane = gtid & 31;
  if (wave >= BB * HH * TT) return;
  float* __restrict__ p = scores + (size_t)wave * SS;

  float mx = -3.402823466e38f;
  for (int i = lane; i < SS; i += 32) mx = fmaxf(mx, p[i]);
  for (int o = 16; o > 0; o >>= 1) mx = fmaxf(mx, __shfl_xor(mx, o, 32));

  float sum = 0.f;
  for (int i = lane; i < SS; i += 32) {
    float e = __expf(p[i] - mx);
    p[i] = e;
    sum += e;
  }
  for (int o = 16; o > 0; o >>= 1) sum += __shfl_xor(sum, o, 32);

  const float inv = 1.0f / sum;
  for (int i = lane; i < SS; i += 32) p[i] *= inv;
}

// ---------------------------------------------------------------------------
// AO[(b*T+t)][h*128 + d] = sum_s attn[bh][t][s] * V'[b,h,s,d]
// One wave per (b,h, d-tile of 16); K-loop over S=2064.
// ---------------------------------------------------------------------------
__global__ void av_wmma(const float* __restrict__ scores,
                        const float* __restrict__ newV,
                        float* __restrict__ AO) {
  const int gtid = blockIdx.x * blockDim.x + threadIdx.x;
  const int wave = gtid >> 5;
  const int lane = gtid & 31;
  const int nt = wave & 7;               // 8 d-tiles
  const int bh = wave >> 3;
  if (bh >= BB * HH) return;
  const int b = bh >> 4, h = bh & 15;
  const int half = lane >> 4, lm = lane & 15;

  const float* __restrict__ arow = scores + (size_t)bh * TT * SS +
                                   (size_t)lm * SS + half * 2;
  const float* __restrict__ vbase = newV + (size_t)b * C_BSTR +
                                    (size_t)h * C_HSTR + nt * 16 + lm;

  v8f acc = {};
  for (int k0 = 0; k0 < SS; k0 += 4) {
    v2f a = *(const v2f*)(arow + k0);
    const int ks = k0 + half * 2;
    v2f bb;
    bb.x = vbase[(size_t)ks * DD];
    bb.y = vbase[(size_t)(ks + 1) * DD];
    acc = __builtin_amdgcn_wmma_f32_16x16x4_f32(false, a, false, bb,
                                                (short)0, acc, false, false);
  }
#pragma unroll
  for (int r = 0; r < 8; ++r) {
    AO[(size_t)(b * TT + r + 8 * half) * HID + h * DD + nt * 16 + lm] = acc[r];
  }
}

// ---------------------------------------------------------------------------
extern "C" void kernel_launch(void* const* d_in, const int* in_sizes, int n_in,
                              void* d_out, int out_size, void* d_ws, size_t ws_size,
                              hipStream_t stream) {
  const float* hidden = (const float*)d_in[0];
  const float* cachek = (const float*)d_in[1];
  const float* cachev = (const float*)d_in[2];
  const float* Wq     = (const float*)d_in[3];
  const float* Wk     = (const float*)d_in[4];
  const float* Wv     = (const float*)d_in[5];
  const float* Wo     = (const float*)d_in[6];

  float* out   = (float*)d_out;                            // (B,T,HID)
  const size_t cacheElems = (size_t)BB * HH * SMAX * DD;   // 33554432
  float* newK = out + (size_t)MM * HID;                    // new_cache_k
  float* newV = newK + cacheElems;                         // new_cache_v

  // workspace layout (floats)
  float* ws      = (float*)d_ws;
  float* ws_q    = ws;                         // 131072
  float* ws_k    = ws_q + (size_t)MM * HID;    // 131072
  float* ws_v    = ws_k + (size_t)MM * HID;    // 131072
  float* ws_ao   = ws_v + (size_t)MM * HID;    // 131072
  float* ws_sc   = ws_ao + (size_t)MM * HID;   // 64*16*2064 = 2113536

  // 1) stream the unchanged cache into the output (memory-bound bulk of work)
  hipMemcpyAsync(newK, cachek, cacheElems * sizeof(float),
                 hipMemcpyDeviceToDevice, stream);
  hipMemcpyAsync(newV, cachev, cacheElems * sizeof(float),
                 hipMemcpyDeviceToDevice, stream);

  // 2) QKV projections: 4 mtiles * 128 ntiles = 512 waves each
  {
    const int waves = (MM / 16) * (HID / 16);
    const int block = 128, grid = (waves * 32 + block - 1) / block;
    gemm_xwt_wmma<<<grid, block, 0, stream>>>(hidden, Wq, ws_q, HID, HID);
    gemm_xwt_wmma<<<grid, block, 0, stream>>>(hidden, Wk, ws_k, HID, HID);
    gemm_xwt_wmma<<<grid, block, 0, stream>>>(hidden, Wv, ws_v, HID, HID);
  }

  // 3) insert new K/V rows at s in [PAST, PAST+T)
  {
    const int n = MM * HID, block = 256;
    scatter_kv<<<(n + block - 1) / block, block, 0, stream>>>(ws_k, ws_v,
                                                              newK, newV);
  }

  // 4) attention scores
  {
    const int waves = BB * HH * (SS / 16);   // 8256
    const int block = 128, grid = (waves * 32 + block - 1) / block;
    qk_scores_wmma<<<grid, block, 0, stream>>>(ws_q, newK, ws_sc);
  }

  // 5) softmax
  {
    const int waves = BB * HH * TT;          // 1024
    const int block = 128, grid = (waves * 32 + block - 1) / block;
    softmax_rows<<<grid, block, 0, stream>>>(ws_sc);
  }

  // 6) attn @ V
  {
    const int waves = BB * HH * (DD / 16);   // 512
    const int block = 128, grid = (waves * 32 + block - 1) / block;
    av_wmma<<<grid, block, 0, stream>>>(ws_sc, newV, ws_ao);
  }

  // 7) output projection
  {
    const int waves = (MM / 16) * (HID / 16);
    const int block = 128, grid = (waves * 32 + block - 1) / block;
    gemm_xwt_wmma<<<grid, block, 0, stream>>>(ws_ao, Wo, out, HID, HID);
  }
}